// PrototypeNetwork_13142599925900
// MI455X (gfx1250) — compile-verified
//
#include <hip/hip_runtime.h>

#define NUM_CLASSES 100
#define FDIM        128
#define EPS         1e-8f

constexpr int NBLK      = 256;                   // streaming blocks (passes A & C)
constexpr int WAVES     = 4;                     // waves per streaming block
constexpr int ACC_ELEMS = NUM_CLASSES * FDIM;    // 12800 floats per accumulator
constexpr int NTILES    = ACC_ELEMS / 16;        // 800 16-element tiles

typedef float v2f __attribute__((ext_vector_type(2)));
typedef float v8f __attribute__((ext_vector_type(8)));

// ---------------------------------------------------------------- init counts
__global__ void zero_counts_kernel(int* __restrict__ counts) {
    int c = threadIdx.x;
    if (c < NUM_CLASSES) counts[c] = 0;
}

// ---------------------------------------------------------------- pass A
// One wave per sample at a time: float4/lane covers 128 dims, shfl-xor reduce
// for the norm. Each wave owns a private LDS accumulator (deterministic order),
// merged into one per-block partial at the end.
__global__ __launch_bounds__(128) void pass_a_kernel(
    const float* __restrict__ f, const int* __restrict__ label,
    float* __restrict__ rinv, float* __restrict__ Pm,
    int* __restrict__ counts, int N)
{
    __shared__ float acc[WAVES * ACC_ELEMS];   // 204,800 bytes
    const int tid  = threadIdx.x;
    const int wave = tid >> 5;
    const int lane = tid & 31;

    for (int e = tid; e < WAVES * ACC_ELEMS; e += blockDim.x) acc[e] = 0.0f;
    __syncthreads();

    const int NW  = gridDim.x * WAVES;
    const int gw  = blockIdx.x * WAVES + wave;
    const int per = (N + NW - 1) / NW;
    const int s0  = gw * per;
    const int s1  = (s0 + per < N) ? (s0 + per) : N;
    float* wacc   = acc + wave * ACC_ELEMS;

    for (int s = s0; s < s1; ++s) {
        float4 v = reinterpret_cast<const float4*>(f + (size_t)s * FDIM)[lane];
        float ss = v.x * v.x + v.y * v.y + v.z * v.z + v.w * v.w;
        #pragma unroll
        for (int off = 16; off > 0; off >>= 1) ss += __shfl_xor(ss, off, 32);
        float r = 1.0f / fmaxf(sqrtf(ss), EPS);
        int   c = label[s];
        if (lane == 0) {
            rinv[s] = r;
            atomicAdd(&counts[c], 1);           // integer add: deterministic
        }
        float* a = wacc + c * FDIM + lane * 4;
        float4 o = *reinterpret_cast<float4*>(a);
        o.x += v.x * r; o.y += v.y * r; o.z += v.z * r; o.w += v.w * r;
        *reinterpret_cast<float4*>(a) = o;
    }
    __syncthreads();

    for (int e = tid; e < ACC_ELEMS; e += blockDim.x) {
        float ssum = acc[e];
        ssum += acc[ACC_ELEMS + e];
        ssum += acc[2 * ACC_ELEMS + e];
        ssum += acc[3 * ACC_ELEMS + e];
        Pm[(size_t)blockIdx.x * ACC_ELEMS + e] = ssum;
    }
}

// ---------------------------------------------------------------- counts -> coeffs
__global__ void finalize_counts_kernel(const int* __restrict__ counts,
                                       float* __restrict__ wden,
                                       float* __restrict__ invn)
{
    int c = threadIdx.x;
    if (c < NUM_CLASSES) {
        float n = (float)counts[c];
        wden[c] = 1.0f / (fmaxf(n - 1.0f, 1.0f) * fmaxf(n, 1.0f));
        invn[c] = 1.0f / fmaxf(n, 1.0f);
    }
}

// ---------------------------------------------------------------- WMMA reduce
// Sum P[blk][e] over blk for one 16-element tile per wave using chained
// V_WMMA_F32_16X16X4_F32 with B == ones:  D[m,n] = sum_k A[m,k] + C[m,n].
// A 16x4 f32 layout: lanes 0-15 hold K=0 (v0) / K=1 (v1); lanes 16-31 K=2/K=3.
// C/D 16x16 f32 layout: lanes 0-15 -> M=r, lanes 16-31 -> M=8+r (N = lane&15).
__global__ __launch_bounds__(32) void reduce_wmma_kernel(
    const float* __restrict__ P, float* __restrict__ outv,
    const float* __restrict__ scale_per_class, int nblk)
{
    const int lane = threadIdx.x;
    const int tile = blockIdx.x;
    const int mrow = lane & 15;
    const int hi   = lane >> 4;

    v8f c = {};
    v2f b; b.x = 1.0f; b.y = 1.0f;
    const float* base = P + tile * 16 + mrow;

    for (int kb = 0; kb < nblk; kb += 4) {
        int k0 = kb + hi * 2;
        v2f a;
        a.x = base[(size_t)k0 * ACC_ELEMS];
        a.y = base[(size_t)(k0 + 1) * ACC_ELEMS];
        c = __builtin_amdgcn_wmma_f32_16x16x4_f32(
                false, a, false, b, (short)0, c, false, false);
    }

    if (mrow == 0) {                       // lanes 0 and 16 write 8 rows each
        int mbase = hi * 8;
        #pragma unroll
        for (int r = 0; r < 8; ++r) {
            int e = tile * 16 + mbase + r;
            float val = c[r];
            if (scale_per_class) val *= scale_per_class[e / FDIM];
            outv[e] = val;
        }
    }
}

// ---------------------------------------------------------------- pass C
// Shared read-only m (51.2KB) + per-wave private proto accumulators (204.8KB).
// Per sample: u = f*rinv, t = u.m[label] - u.u reduced across the wave,
// w = t * wden[label], accumulate w*f.
__global__ __launch_bounds__(128) void pass_c_kernel(
    const float* __restrict__ f, const int* __restrict__ label,
    const float* __restrict__ rinv, const float* __restrict__ m,
    const float* __restrict__ wden, float* __restrict__ Pp, int N)
{
    __shared__ float sm[ACC_ELEMS];            // 51,200 B
    __shared__ float sden[NUM_CLASSES];
    __shared__ float acc[WAVES * ACC_ELEMS];   // 204,800 B  (total ~256 KB)

    const int tid  = threadIdx.x;
    const int wave = tid >> 5;
    const int lane = tid & 31;

    for (int e = tid; e < ACC_ELEMS; e += blockDim.x) sm[e] = m[e];
    if (tid < NUM_CLASSES) sden[tid] = wden[tid];
    for (int e = tid; e < WAVES * ACC_ELEMS; e += blockDim.x) acc[e] = 0.0f;
    __syncthreads();

    const int NW  = gridDim.x * WAVES;
    const int gw  = blockIdx.x * WAVES + wave;
    const int per = (N + NW - 1) / NW;
    const int s0  = gw * per;
    const int s1  = (s0 + per < N) ? (s0 + per) : N;
    float* wacc   = acc + wave * ACC_ELEMS;

    for (int s = s0; s < s1; ++s) {
        float4 v = reinterpret_cast<const float4*>(f + (size_t)s * FDIM)[lane];
        float  r = rinv[s];
        int    c = label[s];
        float4 mm = *reinterpret_cast<const float4*>(sm + c * FDIM + lane * 4);
        float ux = v.x * r, uy = v.y * r, uz = v.z * r, uw = v.w * r;
        float t = ux * mm.x + uy * mm.y + uz * mm.z + uw * mm.w
                - (ux * ux + uy * uy + uz * uz + uw * uw);
        #pragma unroll
        for (int off = 16; off > 0; off >>= 1) t += __shfl_xor(t, off, 32);
        float w = t * sden[c];

        float* a = wacc + c * FDIM + lane * 4;
        float4 o = *reinterpret_cast<float4*>(a);
        o.x += w * v.x; o.y += w * v.y; o.z += w * v.z; o.w += w * v.w;
        *reinterpret_cast<float4*>(a) = o;
    }
    __syncthreads();

    for (int e = tid; e < ACC_ELEMS; e += blockDim.x) {
        float ssum = acc[e];
        ssum += acc[ACC_ELEMS + e];
        ssum += acc[2 * ACC_ELEMS + e];
        ssum += acc[3 * ACC_ELEMS + e];
        Pp[(size_t)blockIdx.x * ACC_ELEMS + e] = ssum;
    }
}

// ---------------------------------------------------------------- launch
extern "C" void kernel_launch(void* const* d_in, const int* in_sizes, int n_in,
                              void* d_out, int out_size, void* d_ws, size_t ws_size,
                              hipStream_t stream)
{
    const float* f     = (const float*)d_in[0];   // [N,128] f32
    const int*   label = (const int*)d_in[1];     // [N] i32
    const int    N     = in_sizes[1];
    float*       out   = (float*)d_out;           // [100,128] f32

    // workspace layout (all f32 unless noted)
    char* w = (char*)d_ws;
    float* rinv    = (float*)w;  w += sizeof(float) * (size_t)N;
    float* partial = (float*)w;  w += sizeof(float) * (size_t)NBLK * ACC_ELEMS;
    float* m       = (float*)w;  w += sizeof(float) * ACC_ELEMS;
    float* wden    = (float*)w;  w += sizeof(float) * NUM_CLASSES;
    float* invn    = (float*)w;  w += sizeof(float) * NUM_CLASSES;
    int*   counts  = (int*)w;

    zero_counts_kernel<<<1, 128, 0, stream>>>(counts);
    pass_a_kernel<<<NBLK, 128, 0, stream>>>(f, label, rinv, partial, counts, N);
    reduce_wmma_kernel<<<NTILES, 32, 0, stream>>>(partial, m, nullptr, NBLK);
    finalize_counts_kernel<<<1, 128, 0, stream>>>(counts, wden, invn);
    pass_c_kernel<<<NBLK, 128, 0, stream>>>(f, label, rinv, m, wden, partial, N);
    reduce_wmma_kernel<<<NTILES, 32, 0, stream>>>(partial, out, invn, NBLK);
}